// GCN_5497558138994
// MI455X (gfx1250) — compile-verified
//
#include <hip/hip_runtime.h>
#include <hip/hip_bf16.h>
#include <cstdint>

// ---------------------------------------------------------------------------
// GCN (3-layer) for MI455X / gfx1250, wave32.
// GEMMs use V_WMMA_F32_16X16X4_F32 (native fp32 WMMA) since the workload is
// memory/atomic bound (x read = 200MB HBM; gather/scatter is L2-resident:
// h/agg are 25.6MB each << 192MB L2). fp32 keeps reference precision.
// Weight tiles are staged to LDS with GLOBAL_LOAD_ASYNC_TO_LDS (ASYNCcnt).
// ---------------------------------------------------------------------------

typedef float v2f __attribute__((ext_vector_type(2)));
typedef float v8f __attribute__((ext_vector_type(8)));

#define LDB    80   // LDS row stride for W tile: 2*80 mod 64 = 32 -> the two
                    // half-wave B reads land on disjoint bank groups
#define KCHUNK 100  // K-chunk staged in LDS (100*80*4 = 32KB static LDS)

// ---- async global->LDS staging (CDNA5 GLOBAL_LOAD_ASYNC_TO_LDS_B64) -------
#if defined(__has_builtin)
#if __has_builtin(__builtin_amdgcn_global_load_async_to_lds_b64) && \
    __has_builtin(__builtin_amdgcn_s_wait_asynccnt)
#define HAVE_ASYNC_LDS 1
#endif
#endif
#ifndef HAVE_ASYNC_LDS
#define HAVE_ASYNC_LDS 0
#warning "gfx1250 async-to-LDS builtins unavailable: falling back to VGPR staging"
#endif

#if HAVE_ASYNC_LDS
// Parameter type revealed by clang diagnostic: pointer to 2 x i32 vector.
typedef int v2i __attribute__((vector_size(8)));
typedef __attribute__((address_space(1))) v2i GV2I;   // global src
typedef __attribute__((address_space(3))) v2i LV2I;   // LDS dst
#endif

// ---------------------------- utility kernels ------------------------------

__global__ void k_zero(float* __restrict__ p, long long n) {
  long long i = (long long)blockIdx.x * blockDim.x + threadIdx.x;
  long long stride = (long long)gridDim.x * blockDim.x;
  for (; i < n; i += stride) p[i] = 0.0f;
}

__global__ void k_deg(const long long* __restrict__ dst, float* __restrict__ deg, int E) {
  int e = blockIdx.x * blockDim.x + threadIdx.x;
  if (e < E) atomicAdd(&deg[dst[e]], 1.0f);
}

__global__ void k_rsqrt_inplace(float* __restrict__ deg, int N) {
  int i = blockIdx.x * blockDim.x + threadIdx.x;
  if (i < N) deg[i] = rsqrtf(deg[i] + 1.0f);
}

// ------------------------------ WMMA GEMM ----------------------------------
// C[M x ldc] = A[M x Kdim] @ W[Kdim x wcols]  (pad cols hold garbage; store
// guarded to storeCols, so garbage never escapes).
// Block: 256 thr = 8 waves -> 32 rows x 64 cols of C per block.
__global__ __launch_bounds__(256) void k_gemm_wmma(
    const float* __restrict__ A, const float* __restrict__ W,
    float* __restrict__ C, int M, int Kdim, int wcols, int ldc, int storeCols) {
  __shared__ float ldsW[KCHUNK * LDB];

  const int tid  = threadIdx.x;
  const int lane = tid & 31;
  const int half = lane >> 4;        // 0: K even pair, 1: K odd pair (per ISA A/B layout)
  const int mrow = lane & 15;
  const int wave = tid >> 5;         // 0..7
  const int colT = (wave & 3) * 16;  // 0,16,32,48
  const int rowT = blockIdx.x * 32 + (wave >> 2) * 16;

  const int arowIdx = min(rowT + mrow, M - 1);
  const float* arow = A + (size_t)arowIdx * Kdim;

  v8f acc = {};

  const int halfW = wcols >> 1;      // W rows handled as float2 pairs

  for (int kc = 0; kc < Kdim; kc += KCHUNK) {
    const int kl = min(KCHUNK, Kdim - kc);
    __syncthreads();                 // previous chunk fully consumed
#if HAVE_ASYNC_LDS
    // DMA the K-chunk of W straight into LDS (no VGPR roundtrip); each lane
    // moves one 8B pair per issue. LDS row stride LDB=80 keeps 8B alignment
    // (c even) and de-conflicts banks for the two half-wave B reads.
    for (int i = tid; i < kl * halfW; i += 256) {
      int k = i / halfW;
      int c = (i - k * halfW) * 2;
      __builtin_amdgcn_global_load_async_to_lds_b64(
          (GV2I*)(W + (size_t)(kc + k) * wcols + c),
          (LV2I*)(ldsW + k * LDB + c),
          0, 0);
    }
    __builtin_amdgcn_s_wait_asynccnt(0);
#else
    for (int i = tid; i < kl * halfW; i += 256) {
      int k = i / halfW;
      int c = (i - k * halfW) * 2;
      *(v2f*)(ldsW + k * LDB + c) = *(const v2f*)(W + (size_t)(kc + k) * wcols + c);
    }
#endif
    __syncthreads();

#pragma unroll 5
    for (int k0 = 0; k0 < kl; k0 += 4) {
      const int koff = k0 + half * 2;
      // A 16x4 fp32 frag: VGPR0 = K=k0(+2 for lanes16-31), VGPR1 = +1
      v2f a = *(const v2f*)(arow + kc + koff);   // 8B aligned: Kdim, kc, koff all even
      // B 4x16 fp32 frag, mirrored layout from LDS
      v2f b;
      b.x = ldsW[koff * LDB + colT + mrow];
      b.y = ldsW[(koff + 1) * LDB + colT + mrow];
      acc = __builtin_amdgcn_wmma_f32_16x16x4_f32(
          false, a, false, b, (short)0, acc, false, false);
    }
  }

  // C layout: VGPR v -> row rowT + v + 8*(lane>=16), col = colT + (lane&15)
#pragma unroll
  for (int v = 0; v < 8; ++v) {
    int row = rowT + v + half * 8;
    int col = colT + mrow;
    if (row < M && col < storeCols)
      C[(size_t)row * ldc + col] = acc[v];
  }
}

// -------------------------- edge scatter (atomics) -------------------------
template <int F>
__global__ __launch_bounds__(256) void k_scatter(
    const float* __restrict__ h, const float* __restrict__ dis,
    const long long* __restrict__ src, const long long* __restrict__ dst,
    float* __restrict__ agg, int E) {
  long long idx = (long long)blockIdx.x * blockDim.x + threadIdx.x;
  if (idx >= (long long)E * F) return;
  int e = (int)(idx / F);
  int f = (int)(idx - (long long)e * F);
  long long s = src[e];
  long long d = dst[e];
  if (f == 0 && e + 4096 < E) {          // warm L2 for upcoming edge ids
    __builtin_prefetch(&src[e + 4096], 0, 1);
    __builtin_prefetch(&dst[e + 4096], 0, 1);
  }
  float norm = dis[s] * dis[d];
  atomicAdd(&agg[(size_t)d * F + f], h[(size_t)s * F + f] * norm);
}

// ----------------- finalize: self-loop + bias (+ BN + ReLU) ----------------
template <int F, bool BN>
__global__ void k_finalize(
    const float* __restrict__ agg, const float* __restrict__ h,
    const float* __restrict__ dis, const float* __restrict__ b,
    const float* __restrict__ g, const float* __restrict__ be,
    const float* __restrict__ m, const float* __restrict__ v,
    float* __restrict__ out, int N) {
  long long idx = (long long)blockIdx.x * blockDim.x + threadIdx.x;
  if (idx >= (long long)N * F) return;
  int i = (int)(idx / F);
  int f = (int)(idx - (long long)i * F);
  float ds = dis[i];
  float val = agg[idx] + h[idx] * ds * ds + b[f];
  if constexpr (BN) {
    val = (val - m[f]) * rsqrtf(v[f] + 1e-5f) * g[f] + be[f];
    val = fmaxf(val, 0.0f);
  }
  out[idx] = val;
}

// ------------------------------ log_softmax --------------------------------
__global__ void k_logsoftmax(const float* __restrict__ logits,
                             float* __restrict__ out, int N, int F) {
  int row  = blockIdx.x * (blockDim.x >> 5) + (threadIdx.x >> 5);
  int lane = threadIdx.x & 31;
  if (row >= N) return;
  const float* r = logits + (size_t)row * F;
  float v0 = (lane < F) ? r[lane] : -1e30f;
  float v1 = (lane + 32 < F) ? r[lane + 32] : -1e30f;
  float mx = fmaxf(v0, v1);
#pragma unroll
  for (int off = 16; off > 0; off >>= 1) mx = fmaxf(mx, __shfl_xor(mx, off, 32));
  float s = 0.0f;
  if (lane < F) s += expf(v0 - mx);
  if (lane + 32 < F) s += expf(v1 - mx);
#pragma unroll
  for (int off = 16; off > 0; off >>= 1) s += __shfl_xor(s, off, 32);
  float ls = logf(s);
  if (lane < F) out[(size_t)row * F + lane] = v0 - mx - ls;
  if (lane + 32 < F) out[(size_t)row * F + lane + 32] = v1 - mx - ls;
}

// ------------------------------- launcher ----------------------------------

extern "C" void kernel_launch(void* const* d_in, const int* in_sizes, int n_in,
                              void* d_out, int out_size, void* d_ws, size_t ws_size,
                              hipStream_t stream) {
  const float*     x   = (const float*)d_in[0];
  const long long* ei  = (const long long*)d_in[1];   // int64 edge_index (2,E)
  const float*     W1  = (const float*)d_in[2];
  const float*     b1  = (const float*)d_in[3];
  const float*     g1  = (const float*)d_in[4];
  const float*     be1 = (const float*)d_in[5];
  const float*     m1  = (const float*)d_in[6];
  const float*     v1  = (const float*)d_in[7];
  const float*     W2  = (const float*)d_in[8];
  const float*     b2  = (const float*)d_in[9];
  const float*     g2  = (const float*)d_in[10];
  const float*     be2 = (const float*)d_in[11];
  const float*     m2  = (const float*)d_in[12];
  const float*     v2  = (const float*)d_in[13];
  const float*     W3  = (const float*)d_in[14];
  const float*     b3  = (const float*)d_in[15];

  const int IN_DIM = 500;
  const int N   = in_sizes[0] / IN_DIM;      // 100000
  const int E   = in_sizes[1] / 2;           // 1600000
  const int HID = in_sizes[2] / IN_DIM;      // 64
  const int OUT = in_sizes[14] / HID;        // 40

  const long long* src = ei;
  const long long* dst = ei + E;

  float* ws  = (float*)d_ws;
  float* dis = ws;                           // N
  float* h   = dis + N;                      // N*HID (layer output of GEMM)
  float* agg = h   + (size_t)N * HID;        // N*HID (scatter accumulator)
  float* p   = agg + (size_t)N * HID;        // N*HID (post-BN activations / logits)

  const int mtiles   = (N + 31) / 32;
  const int eBlocks  = (E + 255) / 256;
  const int nBlocks  = (N + 255) / 256;
  const int ef64     = (int)(((long long)E * 64 + 255) / 256);
  const int ef40     = (int)(((long long)E * 40 + 255) / 256);
  const int nf64     = (int)(((long long)N * 64 + 255) / 256);
  const int nf40     = (int)(((long long)N * 40 + 255) / 256);

  // deg_inv_sqrt = rsqrt(segment_sum(ones, dst) + 1)
  k_zero<<<1024, 256, 0, stream>>>(dis, N);
  k_deg<<<eBlocks, 256, 0, stream>>>(dst, dis, E);
  k_rsqrt_inplace<<<nBlocks, 256, 0, stream>>>(dis, N);

  // ---- layer 1: h = x @ W1 ; agg = scatter ; p = relu(bn(agg + self + b1))
  k_gemm_wmma<<<mtiles, 256, 0, stream>>>(x, W1, h, N, IN_DIM, HID, HID, HID);
  k_zero<<<2048, 256, 0, stream>>>(agg, (long long)N * HID);
  k_scatter<64><<<ef64, 256, 0, stream>>>(h, dis, src, dst, agg, E);
  k_finalize<64, true><<<nf64, 256, 0, stream>>>(agg, h, dis, b1, g1, be1, m1, v1, p, N);

  // ---- layer 2
  k_gemm_wmma<<<mtiles, 256, 0, stream>>>(p, W2, h, N, HID, HID, HID, HID);
  k_zero<<<2048, 256, 0, stream>>>(agg, (long long)N * HID);
  k_scatter<64><<<ef64, 256, 0, stream>>>(h, dis, src, dst, agg, E);
  k_finalize<64, true><<<nf64, 256, 0, stream>>>(agg, h, dis, b2, g2, be2, m2, v2, p, N);

  // ---- layer 3: logits then log_softmax
  k_gemm_wmma<<<mtiles, 256, 0, stream>>>(p, W3, h, N, HID, OUT, OUT, OUT);
  k_zero<<<2048, 256, 0, stream>>>(agg, (long long)N * OUT);
  k_scatter<40><<<ef40, 256, 0, stream>>>(h, dis, src, dst, agg, E);
  k_finalize<40, false><<<nf40, 256, 0, stream>>>(agg, h, dis, b3,
                                                  nullptr, nullptr, nullptr, nullptr, p, N);
  k_logsoftmax<<<(N + 7) / 8, 256, 0, stream>>>(p, (float*)d_out, N, OUT);
}